// Group_47631187312807
// MI455X (gfx1250) — compile-verified
//
#include <hip/hip_runtime.h>
#include <hip/hip_bf16.h>
#include <cstdint>

// Problem constants (reference: xyz [16, 8192, 6] f32, NUM_GROUP=512, GROUP_SIZE=32)
#define BATCH     16
#define NPTS      8192
#define NCH       6
#define NGROUP    512
#define GSIZE     32
#define GTILE     16                 // centers per WMMA tile / per block in kernel B
#define SLAB      512                // points per LDS distance slab in kernel B
#define NB_ELEMS  ((size_t)BATCH * NGROUP * GSIZE * NCH)   // 1,572,864 floats

typedef float v2f __attribute__((ext_vector_type(2)));
typedef float v8f __attribute__((ext_vector_type(8)));

// ---------------------------------------------------------------------------
// Kernel A: furthest point sampling. One block per batch, 512 threads.
// min_d kept in registers (16 contiguous points/thread -> correct lowest-index
// tie-break), coords cached in LDS. Wave32 shuffle argmax + cross-wave merge.
// ---------------------------------------------------------------------------
__global__ __launch_bounds__(512) void fps_kernel(const float* __restrict__ xyz,
                                                  float* __restrict__ centers_ws,
                                                  float* __restrict__ out_center) {
    const int b   = blockIdx.x;
    const int tid = threadIdx.x;
    const int NTH = 512;
    const int PPT = NPTS / NTH;      // 16 points per thread

    __shared__ float sx[NPTS];
    __shared__ float sy[NPTS];
    __shared__ float sz[NPTS];
    __shared__ float red_v[16];
    __shared__ int   red_i[16];
    __shared__ int   s_next;

    for (int i = tid; i < NPTS; i += NTH) {
        const float* p = xyz + ((size_t)b * NPTS + i) * NCH;
        sx[i] = p[0]; sy[i] = p[1]; sz[i] = p[2];
    }
    __syncthreads();

    float mind[PPT];
#pragma unroll
    for (int j = 0; j < PPT; ++j) mind[j] = 1e10f;   // matches reference init

    int last = 0;
    const int base = tid * PPT;

    for (int g = 0; g < NGROUP; ++g) {
        // record center g (= index `last`)
        if (tid == 0) {
            const size_t co = ((size_t)b * NGROUP + g) * 3;
            const float cx = sx[last], cy = sy[last], cz = sz[last];
            centers_ws[co + 0] = cx; centers_ws[co + 1] = cy; centers_ws[co + 2] = cz;
            out_center[co + 0] = cx; out_center[co + 1] = cy; out_center[co + 2] = cz;
        }
        const float lx = sx[last], ly = sy[last], lz = sz[last];

        float bestv = -1.0f;
        int   besti = NPTS;
#pragma unroll
        for (int j = 0; j < PPT; ++j) {
            const int i = base + j;
            const float dx = sx[i] - lx, dy = sy[i] - ly, dz = sz[i] - lz;
            const float d = dx * dx + dy * dy + dz * dz;
            mind[j] = fminf(mind[j], d);
            if (mind[j] > bestv) { bestv = mind[j]; besti = i; }  // strict > keeps lowest idx
        }
        // wave32 reduction (max value, tie -> lower index)
#pragma unroll
        for (int off = 16; off > 0; off >>= 1) {
            const float ov = __shfl_xor(bestv, off, 32);
            const int   oi = __shfl_xor(besti, off, 32);
            if (ov > bestv || (ov == bestv && oi < besti)) { bestv = ov; besti = oi; }
        }
        if ((tid & 31) == 0) { red_v[tid >> 5] = bestv; red_i[tid >> 5] = besti; }
        __syncthreads();
        if (tid == 0) {
            float bv = red_v[0]; int bi = red_i[0];
            for (int w = 1; w < 16; ++w)
                if (red_v[w] > bv || (red_v[w] == bv && red_i[w] < bi)) { bv = red_v[w]; bi = red_i[w]; }
            s_next = bi;
        }
        __syncthreads();
        last = s_next;
    }
}

// ---------------------------------------------------------------------------
// Kernel B: WMMA distance tiles + stable top-32 selection + gather.
// One block per (batch, 16-center tile): grid = 16 * 32 = 512, 256 threads.
// ---------------------------------------------------------------------------
__global__ __launch_bounds__(256) void group_topk_kernel(const float* __restrict__ xyz,
                                                         const float* __restrict__ centers_ws,
                                                         float* __restrict__ out_nb) {
    const int b     = blockIdx.x >> 5;          // / 32
    const int gtile = blockIdx.x & 31;          // % 32
    const int tid   = threadIdx.x;
    const int lane  = tid & 31;
    const int wave  = tid >> 5;                 // 8 waves

    __shared__ float s_dist[GTILE][SLAB];       // 32 KB distance slab
    __shared__ float s_pp[SLAB];                // |p|^2 for slab
    __shared__ float s_cx[GTILE], s_cy[GTILE], s_cz[GTILE], s_cc[GTILE];
    __shared__ float s_topd[GTILE][GSIZE];
    __shared__ int   s_topi[GTILE][GSIZE];

    if (tid < GTILE) {
        const int g = gtile * GTILE + tid;
        const size_t co = ((size_t)b * NGROUP + g) * 3;
        const float cx = centers_ws[co + 0], cy = centers_ws[co + 1], cz = centers_ws[co + 2];
        s_cx[tid] = cx; s_cy[tid] = cy; s_cz[tid] = cz;
        s_cc[tid] = cx * cx + cy * cy + cz * cz;
    }
    for (int t = tid; t < GTILE * GSIZE; t += 256) {
        s_topd[t >> 5][t & 31] = 3.4e38f;
        s_topi[t >> 5][t & 31] = 0;
    }
    __syncthreads();

    const int khalf = lane >> 4;                // 0: K=0,1   1: K=2,3
    const int mlane = lane & 15;

    // A operand (16x4 f32, M x K): lanes 0-15 VGPR{0,1}=K{0,1}; lanes 16-31 VGPR{0,1}=K{2,3}
    v2f a;
    a.x = khalf ? s_cz[mlane] : s_cx[mlane];
    a.y = khalf ? 0.0f        : s_cy[mlane];

    for (int nb0 = 0; nb0 < NPTS; nb0 += SLAB) {
        // |p|^2 for slab
        for (int i = tid; i < SLAB; i += 256) {
            const float* p = xyz + ((size_t)b * NPTS + nb0 + i) * NCH;
            s_pp[i] = p[0] * p[0] + p[1] * p[1] + p[2] * p[2];
        }
        __syncthreads();

        // WMMA distance tiles: each wave computes 4 chunks of 16 points
#pragma unroll
        for (int q = 0; q < 4; ++q) {
            const int ck    = (wave << 2) + q;           // 0..31
            const int n_loc = ck * 16 + mlane;
            const float* p  = xyz + ((size_t)b * NPTS + nb0 + n_loc) * NCH;
            const float px = p[0], py = p[1], pz = p[2];

            // B operand (4x16 f32, K x N): VGPR0 rows K0(l<16)/K1(l>=16); VGPR1 rows K2/K3
            v2f bb;
            bb.x = khalf ? py   : px;
            bb.y = khalf ? 0.0f : pz;

            v8f acc = {};
            acc = __builtin_amdgcn_wmma_f32_16x16x4_f32(
                /*neg_a=*/false, a, /*neg_b=*/false, bb,
                /*c_mod=*/(short)0, acc, /*reuse_a=*/false, /*reuse_b=*/false);

            // D layout: VGPR r, lanes 0-15 -> M=r; lanes 16-31 -> M=r+8; N = lane%16
#pragma unroll
            for (int r = 0; r < 8; ++r) {
                const int mo = r + (khalf << 3);
                s_dist[mo][n_loc] = s_cc[mo] + s_pp[n_loc] - 2.0f * acc[r];
            }
        }
        __syncthreads();

        // Stable streaming top-32 (one thread per center): sorted-insert with
        // strict '<' => ascending by (value, index), matching lax.top_k.
        if (tid < GTILE) {
            float* td = s_topd[tid];
            int*   ti = s_topi[tid];
            for (int j = 0; j < SLAB; ++j) {
                const float dd = s_dist[tid][j];
                if (dd < td[GSIZE - 1]) {
                    const int pidx = nb0 + j;
                    int pos = GSIZE - 1;
                    while (pos > 0 && dd < td[pos - 1]) {
                        td[pos] = td[pos - 1];
                        ti[pos] = ti[pos - 1];
                        --pos;
                    }
                    td[pos] = dd; ti[pos] = pidx;
                }
            }
        }
        __syncthreads();
    }

    // Gather neighborhoods: nb[b, g, k, :] with xyz part re-centered
    for (int t = tid; t < GTILE * GSIZE; t += 256) {
        const int m = t >> 5;
        const int k = t & 31;
        const int g = gtile * GTILE + m;
        const int pi = s_topi[m][k];
        const float* p = xyz + ((size_t)b * NPTS + pi) * NCH;
        float* o = out_nb + (((size_t)b * NGROUP + g) * GSIZE + k) * NCH;
        o[0] = p[0] - s_cx[m];
        o[1] = p[1] - s_cy[m];
        o[2] = p[2] - s_cz[m];
        o[3] = p[3];
        o[4] = p[4];
        o[5] = p[5];
    }
}

extern "C" void kernel_launch(void* const* d_in, const int* in_sizes, int n_in,
                              void* d_out, int out_size, void* d_ws, size_t ws_size,
                              hipStream_t stream) {
    (void)in_sizes; (void)n_in; (void)out_size; (void)ws_size;
    const float* xyz = (const float*)d_in[0];
    float* out = (float*)d_out;
    float* centers_ws = (float*)d_ws;                 // 16*512*3 floats = 96 KB
    float* out_center = out + NB_ELEMS;               // tuple: (nb, center) flat

    fps_kernel<<<BATCH, 512, 0, stream>>>(xyz, centers_ws, out_center);
    group_topk_kernel<<<BATCH * (NGROUP / GTILE), 256, 0, stream>>>(xyz, centers_ws, out);
}